// ContourDec_1022202216666
// MI455X (gfx1250) — compile-verified
//
#include <hip/hip_runtime.h>

typedef __attribute__((ext_vector_type(2))) float v2f;
typedef __attribute__((ext_vector_type(8))) float v8f;

#define SQ2f  1.41421356237309504880f
#define ISQ2f 0.70710678118654752440f

// ------------------------------------------------------------------
// WMMA helper: D = A(16x4) * B(4x16) + C   (fp32, CDNA5 wave32)
// ------------------------------------------------------------------
__device__ __forceinline__ v8f wmma4(v2f a, v2f b, v8f c) {
  return __builtin_amdgcn_wmma_f32_16x16x4_f32(false, a, false, b, (short)0, c,
                                               false, false);
}

// Branch-free banded coefficient: f[t] if 0<=t<K else 0.
// Clamped index keeps the load unconditional (no exec-mask diamonds);
// out-of-band lanes are zeroed with a v_cndmask.
template <int K>
__device__ __forceinline__ float bandcoef(const float* __restrict__ f, int t) {
  int tc = t < 0 ? 0 : (t > K - 1 ? K - 1 : t);
  float v = f[tc];
  return (t == tc) ? v : 0.f;
}

// ------------------------------------------------------------------
// S[n] = sum_c x[n,c]   (512x512 per batch)
// ------------------------------------------------------------------
__global__ void k_chansum(const float* __restrict__ x, float* __restrict__ S, long pb) {
  const int nb  = blockIdx.z;
  const int idx = blockIdx.x * 256 + threadIdx.x;           // 0..262143
  const float* xb = x + (size_t)nb * 8u * 262144u;
  float s = 0.f;
#pragma unroll
  for (int c = 0; c < 8; ++c) s += xb[(size_t)c * 262144u + idx];
  S[(size_t)nb * pb + idx] = s;
}

// ------------------------------------------------------------------
// Horizontal 1-D conv (periodic cols), one 16x16 tile per wave.
// out[r,j] = sum_t f[t] * in[r, (j - cl + t) mod W]
// out = sum_q A_q(16x4) x B_q(4x16):
//   A_q[m,k] = in[r0+m, (w0+4q+k) mod W],  B_q[k,n] = f[(4q+k)-n] (banded)
// A layout (32-bit 16x4): V0 = K0 (lanes 0-15) / K2 (lanes 16-31); V1 = K1/K3.
// B layout (32-bit 4x16): V0 = row K0 (lanes 0-15) / K2 (lanes 16-31); V1 = K1/K3.
// D layout: VGPR p -> row p (lanes 0-15) / p+8 (lanes 16-31), col = lane&15.
// ------------------------------------------------------------------
template <int K>
__global__ void k_rowconv(const float* __restrict__ srcB, float* __restrict__ dstB,
                          const float* __restrict__ f, int cl,
                          int H, int W, long pb) {
  const int lane = threadIdx.x & 31;
  const int half = lane >> 4;        // 0 / 1
  const int m    = lane & 15;        // A row (M), also B/D column (N)
  const int nb   = blockIdx.z;
  const float* S = srcB + (size_t)nb * pb;
  float*       D = dstB + (size_t)nb * pb;
  const int j0 = blockIdx.x * 16, r0 = blockIdx.y * 16;
  const int w0 = j0 - cl;
  constexpr int NQ = (K + 18) >> 2;  // ceil((16+K-1)/4)
  v8f acc = {0.f,0.f,0.f,0.f,0.f,0.f,0.f,0.f};
#pragma unroll
  for (int q = 0; q < NQ; ++q) {
    const int s0 = 4*q + 2*half;
    v2f a, b;
    const int c0 = (w0 + s0 + W)     & (W - 1);   // W is a power of two
    const int c1 = (w0 + s0 + 1 + W) & (W - 1);
    a.x = S[(r0 + m) * W + c0];
    a.y = S[(r0 + m) * W + c1];
    b.x = bandcoef<K>(f, s0 - m);
    b.y = bandcoef<K>(f, s0 + 1 - m);
    acc = wmma4(a, b, acc);
  }
  const int col = j0 + m;
#pragma unroll
  for (int p = 0; p < 8; ++p) D[(r0 + p + 8*half) * W + col] = acc[p];
}

// row-extension sample (per / qper_col half-width rotation), branch-free
template <bool QPER>
__device__ __forceinline__ float colsample(const float* __restrict__ S,
                                           int R, int c, int H, int W) {
  const bool wrapped = (R < 0) | (R >= H);
  R = (R + H) & (H - 1);                       // H is a power of two
  if (QPER) c = wrapped ? ((c + (W >> 1)) & (W - 1)) : c;
  return S[R * W + c];
}

// ------------------------------------------------------------------
// Vertical 1-D conv: out[i,c] = sum_t f[t] * sample(i - ru + t, c)
//   A_q[m,k] = f[(4q+k)-m] (banded),  B_q[k,n] = sample(i0-ru+4q+k, c0+n)
// ------------------------------------------------------------------
template <int K, bool QPER>
__global__ void k_colconv(const float* __restrict__ srcB, float* __restrict__ dstB,
                          const float* __restrict__ f, int ru,
                          int H, int W, long pb) {
  const int lane = threadIdx.x & 31;
  const int half = lane >> 4;
  const int m    = lane & 15;
  const int nb   = blockIdx.z;
  const float* S = srcB + (size_t)nb * pb;
  float*       D = dstB + (size_t)nb * pb;
  const int c0 = blockIdx.x * 16, i0 = blockIdx.y * 16;
  constexpr int NQ = (K + 18) >> 2;
  const int col = c0 + m;
  v8f acc = {0.f,0.f,0.f,0.f,0.f,0.f,0.f,0.f};
#pragma unroll
  for (int q = 0; q < NQ; ++q) {
    const int s0 = 4*q + 2*half;
    v2f a, b;
    a.x = bandcoef<K>(f, s0 - m);
    a.y = bandcoef<K>(f, s0 + 1 - m);
    const int R = i0 - ru + s0;
    b.x = colsample<QPER>(S, R,     col, H, W);
    b.y = colsample<QPER>(S, R + 1, col, H, W);
    acc = wmma4(a, b, acc);
  }
#pragma unroll
  for (int p = 0; p < 8; ++p) D[(i0 + p + 8*half) * W + col] = acc[p];
}

// ------------------------------------------------------------------
// c = avgpool2(xlo); write out0 (broadcast 8 ch); dup8 = 8*upsample(c)
// ------------------------------------------------------------------
__global__ void k_pooldup(const float* __restrict__ xlo, float* __restrict__ dup,
                          float* __restrict__ csm, float* __restrict__ out0, long pb) {
  const int nb = blockIdx.z;
  const int j = threadIdx.x;         // 0..255
  const int i = blockIdx.y;          // 0..255
  const float* A = xlo + (size_t)nb * pb;
  float* Dp = dup + (size_t)nb * pb;
  float* Cs = csm + (size_t)nb * pb;
  const float v = 0.25f * (A[(2*i)*512 + 2*j]     + A[(2*i)*512 + 2*j + 1] +
                           A[(2*i+1)*512 + 2*j]   + A[(2*i+1)*512 + 2*j + 1]);
  Cs[i*256 + j] = v;
  Dp[(2*i)*512 + 2*j]       = 8.f * v;
  Dp[(2*i)*512 + 2*j + 1]   = 0.f;
  Dp[(2*i+1)*512 + 2*j]     = 0.f;
  Dp[(2*i+1)*512 + 2*j + 1] = 0.f;
#pragma unroll
  for (int c = 0; c < 8; ++c)
    out0[((((size_t)nb * 8u + c) * 256u) + i) * 256u + j] = v;
}

// T = S - 8*B  (in place on S)
__global__ void k_sub8(float* __restrict__ a, const float* __restrict__ b, long pb) {
  const int nb = blockIdx.z; const size_t i = (size_t)nb*pb + blockIdx.x*256 + threadIdx.x;
  a[i] -= 8.f * b[i];
}

// level-1 shear gathers of T -> p0sum, p1sum (256x512)
__global__ void k_gatherL1(const float* __restrict__ T, float* __restrict__ p0s,
                           float* __restrict__ p1s, long pb) {
  const int nb = blockIdx.z;
  const int j = blockIdx.x * 256 + threadIdx.x;  // 0..511
  const int i = blockIdx.y;                      // 0..255
  const float* t = T + (size_t)nb * pb;
  const int r = (i - j) & 511;
  p0s[(size_t)nb*pb + i*512 + j] = t[r*512 + ((i + j)     & 511)];
  p1s[(size_t)nb*pb + i*512 + j] = t[r*512 + ((i + j + 1) & 511)];
}

// x0sum = (p0sum - 8*Q1)/sqrt2
__global__ void k_x0sum(const float* __restrict__ p0s, const float* __restrict__ q1,
                        float* __restrict__ o, long pb) {
  const int nb = blockIdx.z; const size_t i = (size_t)nb*pb + blockIdx.x*256 + threadIdx.x;
  o[i] = (p0s[i] - 8.f * q1[i]) * ISQ2f;
}

// U = x0sum + (-sqrt2*p1sum - 8*Q2)
__global__ void k_makeU(const float* __restrict__ x0s, const float* __restrict__ p1s,
                        const float* __restrict__ q2, float* __restrict__ u, long pb) {
  const int nb = blockIdx.z; const size_t i = (size_t)nb*pb + blockIdx.x*256 + threadIdx.x;
  u[i] = x0s[i] + (-SQ2f * p1s[i] - 8.f * q2[i]);
}

// level-2 shear gathers of U -> p0sum2, p1sum2 (256x256)
__global__ void k_gatherL2(const float* __restrict__ U, float* __restrict__ p0s2,
                           float* __restrict__ p1s2, long pb) {
  const int nb = blockIdx.z;
  const int j = threadIdx.x;           // 0..255
  const int i = blockIdx.y;            // 0..255
  const float* u = U + (size_t)nb * pb;
  const int a0 = (i + j) & 255, a1 = (a0 + 1) & 255;
  const int b0 = (2*j - a0) & 511, b1 = (2*j + 1 - a1) & 511;
  p0s2[(size_t)nb*pb + i*256 + j] = u[a0*512 + b0];
  p1s2[(size_t)nb*pb + i*256 + j] = u[a1*512 + b1];
}

// e0sum = (p0sum2 - 16*P1)/sqrt2
__global__ void k_e0sum(const float* __restrict__ p0s2, const float* __restrict__ p1,
                        float* __restrict__ o, long pb) {
  const int nb = blockIdx.z; const size_t i = (size_t)nb*pb + blockIdx.x*256 + threadIdx.x;
  o[i] = (p0s2[i] - 16.f * p1[i]) * ISQ2f;
}

// ------------------------------------------------------------------
// Fused per-channel finisher: e1/e0 for both branches (4 output maps)
// ------------------------------------------------------------------
__global__ void k_final(const float* __restrict__ x, const float* __restrict__ Bf,
                        const float* __restrict__ Q1, const float* __restrict__ Q2,
                        const float* __restrict__ P1, const float* __restrict__ P2,
                        float* __restrict__ out, long pb) {
  const int nb = blockIdx.z >> 3, ch = blockIdx.z & 7;
  const int j = threadIdx.x;           // 0..255
  const int i = blockIdx.x;            // 0..255
  const float* X  = x  + ((size_t)nb * 8u + ch) * 262144u;
  const float* Bb = Bf + (size_t)nb * pb;
  const float* q1 = Q1 + (size_t)nb * pb;
  const float* q2 = Q2 + (size_t)nb * pb;
  const float P1v = P1[(size_t)nb * pb + i*256 + j];
  const float P2v = P2[(size_t)nb * pb + i*256 + j];

  const int a0 = (i + j) & 255, a1 = (a0 + 1) & 255;
  const int b0 = (2*j - a0) & 511, b1 = (2*j + 1 - a1) & 511;

  auto x0f = [&](int a, int b) {
    const int r = (a - b) & 511, c = (a + b) & 511;
    return (X[r*512 + c] - Bb[r*512 + c] - q1[a*512 + b]) * ISQ2f;
  };
  auto x1f = [&](int a, int b) {
    const int r = (a - b) & 511, c = (a + b + 1) & 511;
    return -SQ2f * (X[r*512 + c] - Bb[r*512 + c]) - q2[a*512 + b];
  };

  const size_t MS = 4194304u;          // elements per output map
  const size_t ob = (((size_t)nb * 8u + ch) * 256u + i) * 256u + j;
  out[MS*1 + ob] = -SQ2f * x0f(a1, b1) - P2v;       // e1[:, :8]
  out[MS*2 + ob] = (x0f(a0, b0) - P1v) * ISQ2f;     // e0[:, :8]
  out[MS*3 + ob] = -SQ2f * x1f(a1, b1) - P2v;       // e1[:, 8:]
  out[MS*4 + ob] = (x1f(a0, b0) - P1v) * ISQ2f;     // e0[:, 8:]
}

// ------------------------------------------------------------------
extern "C" void kernel_launch(void* const* d_in, const int* in_sizes, int n_in,
                              void* d_out, int out_size, void* d_ws, size_t ws_size,
                              hipStream_t stream) {
  const float* x = (const float*)d_in[0];
  const float* h = (const float*)d_in[1];   // 9 taps
  const float* g = (const float*)d_in[2];   // 7 taps
  const float* f = (const float*)d_in[3];   // 12 taps
  float* out = (float*)d_out;
  float* ws  = (float*)d_ws;

  // Per-batch workspace layout (floats). PB*8*4B ~= 46 MB total.
  const long PB = 1507328;
  float* F0 = ws + 0;        // S -> T            (512x512)
  float* F1 = ws + 262144;   // t1 / dup8 / B     (512x512)
  float* F2 = ws + 524288;   // xlo / t2 ; then p0sum | p1sum ; then U in lo half
  float* F3 = ws + 786432;   // conv scratch t3/t4/t5/t6 (256x512)
  float* Q1 = ws + 917504;   // (256x512)
  float* Q2 = ws + 1048576;  // (256x512)
  float* F5 = ws + 1179648;  // x0sum ; then p0sum2 | p1sum2/e0sum
  float* CS = ws + 1310720;  // c_small (256x256)
  float* P1 = ws + 1376256;  // (256x256)
  float* P2 = ws + 1441792;  // (256x256)

  const dim3 B256(256), B32(32);
  const dim3 G512c(32, 32, 8);   // 512x512 conv tiles
  const dim3 G2512c(32, 16, 8);  // 256x512 conv tiles
  const dim3 G256c(16, 16, 8);   // 256x256 conv tiles

  // ---- laplacian pyramid (shared fields) ----
  k_chansum<<<dim3(1024,1,8), B256, 0, stream>>>(x, F0, PB);
  k_rowconv<9><<<G512c, B32, 0, stream>>>(F0, F1, h, 4, 512, 512, PB);
  k_colconv<9,false><<<G512c, B32, 0, stream>>>(F1, F2, h, 4, 512, 512, PB);
  k_pooldup<<<dim3(1,256,8), B256, 0, stream>>>(F2, F1, CS, out, PB);
  k_rowconv<7><<<G512c, B32, 0, stream>>>(F1, F2, g, 3, 512, 512, PB);
  k_colconv<7,false><<<G512c, B32, 0, stream>>>(F2, F1, g, 3, 512, 512, PB); // B -> F1
  k_sub8<<<dim3(1024,1,8), B256, 0, stream>>>(F0, F1, PB);                   // T = S - 8B

  // ---- DFB level 1 (channel sums) ----
  k_gatherL1<<<dim3(2,256,8), B256, 0, stream>>>(F0, F2, F2 + 131072, PB);
  k_rowconv<12><<<G2512c, B32, 0, stream>>>(F2 + 131072, F3, f, 6, 256, 512, PB);
  k_colconv<12,true><<<G2512c, B32, 0, stream>>>(F3, Q1, f, 6, 256, 512, PB);
  k_x0sum<<<dim3(512,1,8), B256, 0, stream>>>(F2, Q1, F5, PB);
  k_rowconv<12><<<G2512c, B32, 0, stream>>>(F5, F3, f, 5, 256, 512, PB);
  k_colconv<12,true><<<G2512c, B32, 0, stream>>>(F3, Q2, f, 5, 256, 512, PB);
  k_makeU<<<dim3(512,1,8), B256, 0, stream>>>(F5, F2 + 131072, Q2, F2, PB);

  // ---- DFB level 2 (channel sums) ----
  k_gatherL2<<<dim3(1,256,8), B256, 0, stream>>>(F2, F5, F5 + 65536, PB);
  k_rowconv<12><<<G256c, B32, 0, stream>>>(F5 + 65536, F3, f, 6, 256, 256, PB);
  k_colconv<12,false><<<G256c, B32, 0, stream>>>(F3, P1, f, 6, 256, 256, PB);
  k_e0sum<<<dim3(256,1,8), B256, 0, stream>>>(F5, P1, F5 + 65536, PB);
  k_rowconv<12><<<G256c, B32, 0, stream>>>(F5 + 65536, F3, f, 5, 256, 256, PB);
  k_colconv<12,false><<<G256c, B32, 0, stream>>>(F3, P2, f, 5, 256, 256, PB);

  // ---- fused per-channel outputs (bandwidth-bound main pass) ----
  k_final<<<dim3(256,1,64), B256, 0, stream>>>(x, F1, Q1, Q2, P1, P2, out, PB);
}